// RGBLambertianRendererWithVisibility_3624952398724
// MI455X (gfx1250) — compile-verified
//
#include <hip/hip_runtime.h>
#include <math.h>

#define WPB 8   // waves per 256-thread block
#define SPW 8   // consecutive samples processed per wave (ray-run aggregation)

// gfx1250 async global->LDS path, guarded so the file always compiles.
#if defined(__gfx1250__) && __has_builtin(__builtin_amdgcn_global_load_async_to_lds_b128) && __has_builtin(__builtin_amdgcn_s_wait_asynccnt)
#define USE_ASYNC 1
#else
#define USE_ASYNC 0
#endif

__device__ __forceinline__ float wave_sum32(float v) {
#pragma unroll
  for (int m = 16; m > 0; m >>= 1) v += __shfl_xor(v, m, 32);
  return v;
}

__device__ __forceinline__ float srgb1(float x) {
  float safe = fmaxf(x, 1e-8f);
  float hi = 1.055f * powf(safe, 0.41666666666666667f) - 0.055f;
  return (x <= 0.0031308f) ? (12.92f * x) : hi;
}

#if USE_ASYNC
// The builtin's params are pointers to 16-byte int vectors (per hipcc diagnostic):
//   (v4i __device__* gsrc, v4i __shared__* ldsdst, imm offset, imm cpol)
typedef int v4i __attribute__((vector_size(16)));
typedef __attribute__((address_space(1))) v4i as1_v4i;
typedef __attribute__((address_space(3))) v4i as3_v4i;

// Copy 768 contiguous bytes global->LDS using async b128 copies (16B/lane).
__device__ __forceinline__ void stage768(void* lds_dst, const void* g_src, int lane) {
  char* l = (char*)lds_dst + lane * 16;
  const char* g = (const char*)g_src + lane * 16;
  __builtin_amdgcn_global_load_async_to_lds_b128((as1_v4i*)g, (as3_v4i*)l, 0, 0);
  if (lane < 16)
    __builtin_amdgcn_global_load_async_to_lds_b128((as1_v4i*)(g + 512),
                                                   (as3_v4i*)(l + 512), 0, 0);
}
#endif

__global__ void __launch_bounds__(WPB * 32)
zero_ws_kernel(float* p, int n) {
  int i = blockIdx.x * blockDim.x + threadIdx.x;
  if (i < n) p[i] = 0.0f;
}

// Specialized L == 64 kernel: one wave per sample, lane l handles lights l, l+32.
__global__ void __launch_bounds__(WPB * 32)
render_l64(const float* __restrict__ albedos,
           const float* __restrict__ normals,
           const float* __restrict__ ldir,
           const float* __restrict__ lcol,
           const float* __restrict__ vis,
           const float* __restrict__ wts,
           const int*   __restrict__ rays,
           float* __restrict__ comp,
           float* __restrict__ wsum,
           int N)
{
  const int lane = threadIdx.x & 31;
  const int wib  = threadIdx.x >> 5;
  const int wave = __builtin_amdgcn_readfirstlane((int)(blockIdx.x * WPB + wib));
  const long s0 = (long)wave * SPW;
  if (s0 >= (long)N) return;
  const int nS = (int)(((long)N - s0 < SPW) ? ((long)N - s0) : (long)SPW);

#if USE_ASYNC
  __shared__ alignas(16) float smem[WPB][2][384];   // 2 x (192 dir + 192 color) per wave
  float* bufA = &smem[wib][0][0];
  float* bufB = &smem[wib][1][0];
  stage768(bufA,       ldir + (size_t)s0 * 192, lane);
  stage768(bufA + 192, lcol + (size_t)s0 * 192, lane);
  int pp = 0;
#endif

  int curRay = -1;
  float aR = 0.f, aG = 0.f, aB = 0.f, aW = 0.f;

  for (int k = 0; k < nS; ++k) {
    const long s = s0 + k;
#if USE_ASYNC
    float* cur = pp ? bufB : bufA;
    if (k + 1 < nS) {
      float* nxt = pp ? bufA : bufB;
      stage768(nxt,       ldir + (size_t)(s + 1) * 192, lane);
      stage768(nxt + 192, lcol + (size_t)(s + 1) * 192, lane);
      __builtin_amdgcn_s_wait_asynccnt(4);   // current buffer's 4 copies done
    } else {
      __builtin_amdgcn_s_wait_asynccnt(0);
    }
    __asm__ volatile("" ::: "memory");
    const float* ld = cur;
    const float* lc = cur + 192;
    pp ^= 1;
#else
    const float* ld = ldir + (size_t)s * 192;
    const float* lc = lcol + (size_t)s * 192;
    if (k + 1 < nS) {   // global_prefetch_b8 for next sample's tiles
      __builtin_prefetch(ldir + (size_t)(s + 1) * 192 + lane * 6, 0, 1);
      __builtin_prefetch(lcol + (size_t)(s + 1) * 192 + lane * 6, 0, 1);
    }
#endif
    const float* vv = vis + (size_t)s * 64;
    const float nx = normals[s * 3 + 0];
    const float ny = normals[s * 3 + 1];
    const float nz = normals[s * 3 + 2];

    float c = 0.f, sr = 0.f, sg = 0.f, sb = 0.f;
#pragma unroll
    for (int h = 0; h < 2; ++h) {
      const int j = lane + h * 32;
      const float dx = ld[j * 3 + 0];
      const float dy = ld[j * 3 + 1];
      const float dz = ld[j * 3 + 2];
      const float d  = fmaf(nx, dx, fmaf(ny, dy, nz * dz));
      c += (d > 0.f) ? 1.f : 0.f;
      const float t = fminf(fmaxf(d, 0.f), 1.f) * vv[j];
      sr = fmaf(t, lc[j * 3 + 0], sr);
      sg = fmaf(t, lc[j * 3 + 1], sg);
      sb = fmaf(t, lc[j * 3 + 2], sb);
    }
    c  = wave_sum32(c);
    sr = wave_sum32(sr);
    sg = wave_sum32(sg);
    sb = wave_sum32(sb);
    c  = fmaxf(c, 1.f);                 // where(count>0, count, 1)

    const float w  = wts[s];
    const float sc = w / c;             // fold weight and 1/count
    const float cr = albedos[s * 3 + 0] * sr * sc;
    const float cg = albedos[s * 3 + 1] * sg * sc;
    const float cb = albedos[s * 3 + 2] * sb * sc;

    const int ray = rays[s];
    if (ray != curRay) {                // sorted runs -> aggregate, flush on change
      if (curRay >= 0 && lane == 0) {
        atomicAdd(&comp[curRay * 3 + 0], aR);
        atomicAdd(&comp[curRay * 3 + 1], aG);
        atomicAdd(&comp[curRay * 3 + 2], aB);
        atomicAdd(&wsum[curRay], aW);
      }
      curRay = ray;
      aR = aG = aB = aW = 0.f;
    }
    aR += cr; aG += cg; aB += cb; aW += w;
  }
  if (curRay >= 0 && lane == 0) {
    atomicAdd(&comp[curRay * 3 + 0], aR);
    atomicAdd(&comp[curRay * 3 + 1], aG);
    atomicAdd(&comp[curRay * 3 + 2], aB);
    atomicAdd(&wsum[curRay], aW);
  }
}

// Generic-L fallback (direct loads), same algorithm.
__global__ void __launch_bounds__(WPB * 32)
render_gen(const float* __restrict__ albedos,
           const float* __restrict__ normals,
           const float* __restrict__ ldir,
           const float* __restrict__ lcol,
           const float* __restrict__ vis,
           const float* __restrict__ wts,
           const int*   __restrict__ rays,
           float* __restrict__ comp,
           float* __restrict__ wsum,
           int N, int L)
{
  const int lane = threadIdx.x & 31;
  const int wib  = threadIdx.x >> 5;
  const int wave = __builtin_amdgcn_readfirstlane((int)(blockIdx.x * WPB + wib));
  const long s0 = (long)wave * SPW;
  if (s0 >= (long)N) return;
  const int nS = (int)(((long)N - s0 < SPW) ? ((long)N - s0) : (long)SPW);

  int curRay = -1;
  float aR = 0.f, aG = 0.f, aB = 0.f, aW = 0.f;

  for (int k = 0; k < nS; ++k) {
    const long s = s0 + k;
    const float* ld = ldir + (size_t)s * (size_t)L * 3;
    const float* lc = lcol + (size_t)s * (size_t)L * 3;
    const float* vv = vis  + (size_t)s * (size_t)L;
    const float nx = normals[s * 3 + 0];
    const float ny = normals[s * 3 + 1];
    const float nz = normals[s * 3 + 2];

    float c = 0.f, sr = 0.f, sg = 0.f, sb = 0.f;
    for (int j = lane; j < L; j += 32) {
      const float dx = ld[j * 3 + 0];
      const float dy = ld[j * 3 + 1];
      const float dz = ld[j * 3 + 2];
      const float d  = fmaf(nx, dx, fmaf(ny, dy, nz * dz));
      c += (d > 0.f) ? 1.f : 0.f;
      const float t = fminf(fmaxf(d, 0.f), 1.f) * vv[j];
      sr = fmaf(t, lc[j * 3 + 0], sr);
      sg = fmaf(t, lc[j * 3 + 1], sg);
      sb = fmaf(t, lc[j * 3 + 2], sb);
    }
    c  = fmaxf(wave_sum32(c), 1.f);
    sr = wave_sum32(sr);
    sg = wave_sum32(sg);
    sb = wave_sum32(sb);

    const float w  = wts[s];
    const float sc = w / c;
    const float cr = albedos[s * 3 + 0] * sr * sc;
    const float cg = albedos[s * 3 + 1] * sg * sc;
    const float cb = albedos[s * 3 + 2] * sb * sc;

    const int ray = rays[s];
    if (ray != curRay) {
      if (curRay >= 0 && lane == 0) {
        atomicAdd(&comp[curRay * 3 + 0], aR);
        atomicAdd(&comp[curRay * 3 + 1], aG);
        atomicAdd(&comp[curRay * 3 + 2], aB);
        atomicAdd(&wsum[curRay], aW);
      }
      curRay = ray;
      aR = aG = aB = aW = 0.f;
    }
    aR += cr; aG += cg; aB += cb; aW += w;
  }
  if (curRay >= 0 && lane == 0) {
    atomicAdd(&comp[curRay * 3 + 0], aR);
    atomicAdd(&comp[curRay * 3 + 1], aG);
    atomicAdd(&comp[curRay * 3 + 2], aB);
    atomicAdd(&wsum[curRay], aW);
  }
}

__global__ void __launch_bounds__(256)
finalize_kernel(const float* __restrict__ comp,
                const float* __restrict__ wsum,
                const float* __restrict__ bg,
                float* __restrict__ out, int R)
{
  int r = blockIdx.x * blockDim.x + threadIdx.x;
  if (r >= R) return;
  const float f = 1.0f - wsum[r];
  out[r * 3 + 0] = srgb1(fmaf(bg[r * 3 + 0], f, comp[r * 3 + 0]));
  out[r * 3 + 1] = srgb1(fmaf(bg[r * 3 + 1], f, comp[r * 3 + 1]));
  out[r * 3 + 2] = srgb1(fmaf(bg[r * 3 + 2], f, comp[r * 3 + 2]));
}

extern "C" void kernel_launch(void* const* d_in, const int* in_sizes, int n_in,
                              void* d_out, int out_size, void* d_ws, size_t ws_size,
                              hipStream_t stream) {
  const float* albedos = (const float*)d_in[0];
  const float* normals = (const float*)d_in[1];
  const float* ldir    = (const float*)d_in[2];
  const float* lcol    = (const float*)d_in[3];
  const float* vis     = (const float*)d_in[4];
  const float* bg      = (const float*)d_in[5];
  const float* wts     = (const float*)d_in[6];
  const int*   rays    = (const int*)d_in[7];   // JAX default demotes int64 -> int32
  (void)n_in; (void)ws_size; (void)out_size;

  const int N = in_sizes[0] / 3;
  const int R = in_sizes[5] / 3;
  const int L = (N > 0) ? (in_sizes[2] / (3 * N)) : 0;

  float* comp = (float*)d_ws;          // R*3 floats
  float* wsum = comp + (size_t)R * 3;  // R floats

  // zero the accumulators every call (d_ws is not re-poisoned between replays)
  {
    int n = R * 4;
    int blocks = (n + 255) / 256;
    if (blocks > 0)
      zero_ws_kernel<<<blocks, 256, 0, stream>>>(comp, n);
  }

  if (N > 0 && L > 0) {
    const long waves  = ((long)N + SPW - 1) / SPW;
    const int  blocks = (int)((waves + WPB - 1) / WPB);
    if (L == 64) {
      render_l64<<<blocks, WPB * 32, 0, stream>>>(
          albedos, normals, ldir, lcol, vis, wts, rays, comp, wsum, N);
    } else {
      render_gen<<<blocks, WPB * 32, 0, stream>>>(
          albedos, normals, ldir, lcol, vis, wts, rays, comp, wsum, N, L);
    }
  }

  {
    int blocks = (R + 255) / 256;
    if (blocks > 0)
      finalize_kernel<<<blocks, 256, 0, stream>>>(comp, wsum, bg, (float*)d_out, R);
  }
}